// BilateralSolverLocal_30837865185732
// MI455X (gfx1250) — compile-verified
//
#include <hip/hip_runtime.h>

// ---------------------------------------------------------------------------
// Bilateral solver loss on MI455X (gfx1250, wave32).
// HBM-bound: 180 MB of w_ij streamed once  ->  ~7.7 us floor @ 23.3 TB/s.
//  * w_ij streamed with NON-TEMPORAL b128 loads (don't churn the 192MB L2)
//  * guidance image staged in LDS via the Tensor Data Mover (TENSORcnt path)
//  * reductions via v_wmma_f32_16x16x4_f32 (B = ones row-sum trick)
//  * deterministic: no atomics; fixed-order block partials + finalize wave
// ---------------------------------------------------------------------------

#define HH 320
#define WW 320
#define PADR 10
#define NOFF 440

typedef float v2f  __attribute__((ext_vector_type(2)));
typedef float v4f  __attribute__((ext_vector_type(4)));
typedef float v8f  __attribute__((ext_vector_type(8)));
typedef unsigned int u32x4 __attribute__((ext_vector_type(4)));
typedef int i32x4 __attribute__((ext_vector_type(4)));
typedef int i32x8 __attribute__((ext_vector_type(8)));

// ---- wave32 sum via WMMA: A vgpr0 = partials, B = ones(4x16) ---------------
// D[m][n] = sum_k A[m][k]*1 = p_m + p_(m+16)  (A layout: lanes0-15 -> K0/K1,
// lanes16-31 -> K2/K3).  Sum of the 8 C/D vgprs gives rows 0-7 (lanes 0-15)
// and rows 8-15 (lanes 16-31); one xor-16 shuffle completes the reduction.
__device__ __forceinline__ float wave_reduce_sum(float p) {
    v2f a; a[0] = p;    a[1] = 0.0f;
    v2f b; b[0] = 1.0f; b[1] = 1.0f;
    v8f c = {0.f, 0.f, 0.f, 0.f, 0.f, 0.f, 0.f, 0.f};
    c = __builtin_amdgcn_wmma_f32_16x16x4_f32(false, a, false, b,
                                              (short)0, c, false, false);
    float s = c[0] + c[1] + c[2] + c[3] + c[4] + c[5] + c[6] + c[7];
    s += __shfl_xor(s, 16, 32);
    return s;
}

// ---- TDM: async-load nrows x 320 f32 rows from global into LDS -------------
// D# fields per CDNA5 ISA 8.3-8.5 (2-D tensor, data_size=4B, no padding,
// no multicast, no atomic-barrier).
__device__ __forceinline__ void tdm_load_rows(unsigned lds_byte,
                                              const float* gsrc, int nrows) {
    unsigned long long ga = (unsigned long long)(size_t)gsrc;
    u32x4 g0 = {0u, 0u, 0u, 0u};
    g0[0] = 1u;                                            // count = 1 (valid)
    g0[1] = lds_byte;                                      // lds_addr
    g0[2] = (unsigned)(ga & 0xFFFFFFFFull);                // global_addr[31:0]
    g0[3] = ((unsigned)(ga >> 32) & 0x01FFFFFFu)           // global_addr[56:32]
            | 0x80000000u;                                 // type = 2 (image)
    i32x8 g1 = {0, 0, 0, 0, 0, 0, 0, 0};
    g1[0] = 0x00020000;                                    // data_size = 4B
    g1[1] = (int)(WW << 16);                               // tensor_dim0 = 320
    g1[2] = (int)(nrows << 16);                            // tensor_dim1
    g1[3] = (int)(WW << 16);                               // tile_dim0 = 320
    g1[4] = nrows;                                         // tile_dim1 (2-D)
    g1[5] = WW;                                            // tensor_dim0_stride
    i32x4 gz = {0, 0, 0, 0};
#if defined(__clang_major__) && __clang_major__ >= 23
    i32x8 gz8 = {0, 0, 0, 0, 0, 0, 0, 0};
    __builtin_amdgcn_tensor_load_to_lds(g0, g1, gz, gz, gz8, 0);
#else
    __builtin_amdgcn_tensor_load_to_lds(g0, g1, gz, gz, 0);
#endif
}

// ---------------------------------------------------------------------------
// Main kernel: block = (4-row band) x (110-offset chunk).  320 blocks x 320
// threads; thread (r, xg) owns pixels y = yb*4+r, x = 4*xg..4*xg+3 and loops
// its offset chunk, streaming w_ij with non-temporal b128 loads.
// ---------------------------------------------------------------------------
__global__ __launch_bounds__(320)
void bs_smooth(const float* __restrict__ outv, const float* __restrict__ tgt,
               const float* __restrict__ w, float* __restrict__ partial) {
    __shared__ float tile[24 * WW];   // 30.7 KB: rows yb*4-10 .. yb*4+13 (clamped)
    __shared__ float wsum[10];

    const int blk = blockIdx.x;
    const int kc  = blk & 3;          // offset chunk 0..3 (110 offsets each)
    const int yb  = blk >> 2;         // row band 0..79
    const int y0  = yb * 4 - PADR;    // first desired row (may be < 0)
    const int rlo = max(y0, 0);
    const int rhi = min(yb * 4 + 3 + PADR, HH - 1);
    const int nrows = rhi - rlo + 1;
    const int dst = rlo - y0;         // tile row where valid rows start

    if (threadIdx.x < 32) {           // wave 0 issues the DMA and waits on it
        unsigned lds_byte =
            (unsigned)(size_t)(void*)tile + (unsigned)(dst * WW * 4);
        tdm_load_rows(lds_byte, outv + (size_t)rlo * WW, nrows);
        __builtin_amdgcn_s_wait_tensorcnt(0);
    }
    __syncthreads();

    if (nrows < 24) {                 // replicate-pad halo rows inside LDS
        for (int t = threadIdx.x; t < 24 * WW; t += 320) {
            int r = t / WW;
            int x = t - r * WW;
            int src = y0 + r;
            if (src < 0)        tile[t] = tile[dst * WW + x];
            else if (src >= HH) tile[t] = tile[(dst + nrows - 1) * WW + x];
        }
        __syncthreads();
    }

    const int r  = threadIdx.x / 80;          // sub-row 0..3
    const int xg = threadIdx.x - r * 80;      // float4 group 0..79
    const int x  = xg * 4;
    const int y  = yb * 4 + r;

    const v4f cen = *(const v4f*)&tile[(PADR + r) * WW + x];
    const float* wp = w + (size_t)y * WW + x;

    float acc = 0.0f;
    const int k0 = kc * 110;
    const int k1 = k0 + 110;

    // depth-1 software pipeline over the stream-once w_ij: issue next load
    // (non-temporal: keep L2 for the reusable guidance data) while computing.
    v4f wnext = __builtin_nontemporal_load(
        (const v4f*)(wp + (size_t)k0 * (HH * WW)));
    #pragma unroll 5
    for (int k = k0; k < k1; ++k) {
        const v4f wv = wnext;
        if (k + 1 < k1) {
            wnext = __builtin_nontemporal_load(
                (const v4f*)(wp + (size_t)(k + 1) * (HH * WW)));
        }
        const int kl = k + (k >= 220 ? 1 : 0);   // skip center offset (10,10)
        const int i  = kl / 21;
        const int j  = kl - i * 21;
        const float* nrow = &tile[(i + r) * WW];
        const int xb = x + j - PADR;
        const int c0 = min(max(xb,     0), WW - 1);
        const int c1 = min(max(xb + 1, 0), WW - 1);
        const int c2 = min(max(xb + 2, 0), WW - 1);
        const int c3 = min(max(xb + 3, 0), WW - 1);
        const float d0 = cen[0] - nrow[c0];
        const float d1 = cen[1] - nrow[c1];
        const float d2 = cen[2] - nrow[c2];
        const float d3 = cen[3] - nrow[c3];
        acc = fmaf(wv[0] * d0, d0, acc);
        acc = fmaf(wv[1] * d1, d1, acc);
        acc = fmaf(wv[2] * d2, d2, acc);
        acc = fmaf(wv[3] * d3, d3, acc);
    }

    // loss = (LAM/440)*sum(w d^2) + (1/(H*W))*sum((o-t)^2)
    float v = acc * (float)(128.0 / 440.0);
    if (kc == 0) {                    // data term counted exactly once
        const v4f tg = __builtin_nontemporal_load(
            (const v4f*)(tgt + (size_t)y * WW + x));
        const float t0 = cen[0] - tg[0], t1 = cen[1] - tg[1];
        const float t2 = cen[2] - tg[2], t3 = cen[3] - tg[3];
        v += (t0 * t0 + t1 * t1 + t2 * t2 + t3 * t3)
             * (1.0f / (float)(HH * WW));
    }

    const float s = wave_reduce_sum(v);         // v_wmma_f32_16x16x4_f32
    const int wave = threadIdx.x >> 5;
    if ((threadIdx.x & 31) == 0) wsum[wave] = s;
    __syncthreads();
    if (threadIdx.x == 0) {
        float t = 0.0f;
        #pragma unroll
        for (int i2 = 0; i2 < 10; ++i2) t += wsum[i2];
        partial[blk] = t;                       // deterministic (no atomics)
    }
}

// Deterministic final reduction of the 320 block partials (1 wave).
__global__ __launch_bounds__(32)
void bs_finalize(const float* __restrict__ partial, float* __restrict__ res) {
    float p = 0.0f;
    #pragma unroll
    for (int t = 0; t < 10; ++t) p += partial[threadIdx.x + 32 * t];
    const float s = wave_reduce_sum(p);
    if (threadIdx.x == 0) res[0] = s;
}

extern "C" void kernel_launch(void* const* d_in, const int* in_sizes, int n_in,
                              void* d_out, int out_size, void* d_ws,
                              size_t ws_size, hipStream_t stream) {
    const float* outv = (const float*)d_in[0];   // output  [320,320] f32
    const float* tgt  = (const float*)d_in[1];   // target  [320,320] f32
    const float* w    = (const float*)d_in[2];   // w_ij [440,320,320] f32
    float* res = (float*)d_out;                  // scalar f32
    float* partial = (float*)d_ws;               // 320 floats of scratch

    bs_smooth<<<dim3(320), dim3(320), 0, stream>>>(outv, tgt, w, partial);
    bs_finalize<<<dim3(1), dim3(32), 0, stream>>>(partial, res);
}